// Verification_57226144251960
// MI455X (gfx1250) — compile-verified
//
#include <hip/hip_runtime.h>
#include <hip/hip_bf16.h>

typedef __attribute__((ext_vector_type(2))) float v2f;
typedef __attribute__((ext_vector_type(8))) float v8f;

#define B_DIM 64
#define T_DIM 2048
#define H_DIM 512
#define R_DIM (B_DIM * T_DIM)   // 131072 rows

// ---------------------------------------------------------------------------
// Kernel A: fused projection  proj[r][0..3] = {nw0,nw1,w0,w1} + bias
// One wave = one 16-row tile, V_WMMA_F32_16X16X4_F32 accumulating K=512.
// final_x is 256MB read-once (> 192MB L2) -> non-temporal loads.
// ---------------------------------------------------------------------------
__global__ __launch_bounds__(256) void proj_wmma_kernel(
    const float* __restrict__ x,     // [R, 512]
    const float* __restrict__ Wnw,   // [2, 512]
    const float* __restrict__ bnw,   // [2]
    const float* __restrict__ Ww,    // [2, 512]
    const float* __restrict__ bw,    // [2]
    float* __restrict__ proj)        // [R, 4]
{
    const int lane  = threadIdx.x & 31;
    const int wave  = threadIdx.x >> 5;
    const int tile  = blockIdx.x * 8 + wave;        // 16 rows per tile
    const long long rowBase = (long long)tile * 16;

    const int n     = lane & 15;                    // output column (0..3 useful)
    const int khalf = lane >> 4;                    // 0 -> K{0,1}, 1 -> K{2,3}

    // A fragment source: row (rowBase+n), starting at column khalf*2
    const float* arow = x + (rowBase + n) * H_DIM + khalf * 2;

    // B fragment source: weight row for channel n (dummy row + scale 0 for n>=4
    // keeps control flow uniform around the WMMA — EXEC must be all ones).
    const float* wrow;
    float wscale, bias;
    if (n < 2)      { wrow = Wnw + n * H_DIM;       wscale = 1.f; bias = bnw[n];     }
    else if (n < 4) { wrow = Ww  + (n - 2) * H_DIM; wscale = 1.f; bias = bw[n - 2];  }
    else            { wrow = Wnw;                   wscale = 0.f; bias = 0.f;        }

    v8f acc = {};
    #pragma unroll 4
    for (int k = 0; k < H_DIM; k += 4) {
        v2f a = __builtin_nontemporal_load((const v2f*)(arow + k));
        v2f b;
        b.x = wrow[k + khalf * 2 + 0] * wscale;
        b.y = wrow[k + khalf * 2 + 1] * wscale;
        // D(16x16,f32) += A(16x4,f32) * B(4x16,f32)
        acc = __builtin_amdgcn_wmma_f32_16x16x4_f32(
            /*neg_a=*/false, a, /*neg_b=*/false, b,
            /*c_mod=*/(short)0, acc, /*reuse_a=*/false, /*reuse_b=*/false);
    }

    // D element [m][n]: vgpr r holds m = khalf*8 + r on this lane.
    if (n < 4) {
        float* p = proj + (rowBase + khalf * 8) * 4 + n;
        #pragma unroll
        for (int r = 0; r < 8; ++r)
            __builtin_nontemporal_store(acc[r] + bias, p + r * 4);
    }
}

// ---------------------------------------------------------------------------
// Kernel B: masked mix -> noworry_det (output #1, also the p1 LSTM input)
// ---------------------------------------------------------------------------
__global__ __launch_bounds__(256) void mix_kernel(
    const float* __restrict__ proj,          // [R,4]
    const unsigned char* __restrict__ worry, // [R] bool
    float* __restrict__ out_nw)              // [R,2]
{
    const long long i = (long long)blockIdx.x * blockDim.x + threadIdx.x;
    if (i >= R_DIM) return;
    const float4 p = *(const float4*)(proj + i * 4);
    const bool w = worry[i] != 0;
    float2 m;
    m.x = w ? p.z : p.x;
    m.y = w ? p.w : p.y;
    *(float2*)(out_nw + i * 2) = m;
}

// ---------------------------------------------------------------------------
// Kernel C: both tiny LSTMs + adjust_det, one lane per batch element.
// Gate order i,f,g,o (PyTorch). State evolves for all t; only the emitted
// h is masked by lengths (pack/pad semantics). p1: I=2,H=2. p2: I=2,H=1.
// ---------------------------------------------------------------------------
__device__ __forceinline__ float sigf(float x) {
    return 1.f / (1.f + __expf(-x));
}
__device__ __forceinline__ float tanh_fast(float x) {
    float e = __expf(2.f * x);
    return (e - 1.f) / (e + 1.f);
}

__global__ __launch_bounds__(64) void lstm_kernel(
    const float* __restrict__ mixed,   // [B,T,2] = d_out part 1
    const float* __restrict__ proj,    // [B*T,4]; ch 2,3 = worry_det
    const long long* __restrict__ lengths,
    const float* __restrict__ Wih1, const float* __restrict__ Whh1,
    const float* __restrict__ bih1, const float* __restrict__ bhh1,
    const float* __restrict__ Wih2, const float* __restrict__ Whh2,
    const float* __restrict__ bih2, const float* __restrict__ bhh2,
    float* __restrict__ adjust)        // [B,T,2] = d_out part 2
{
    const int b = threadIdx.x;
    if (b >= B_DIM) return;

    float wi1[16], wh1[16], bb1[8];
    #pragma unroll
    for (int i = 0; i < 16; ++i) { wi1[i] = Wih1[i]; wh1[i] = Whh1[i]; }
    #pragma unroll
    for (int i = 0; i < 8; ++i)  bb1[i] = bih1[i] + bhh1[i];
    float wi2[8], wh2[4], bb2[4];
    #pragma unroll
    for (int i = 0; i < 8; ++i)  wi2[i] = Wih2[i];
    #pragma unroll
    for (int i = 0; i < 4; ++i)  { wh2[i] = Whh2[i]; bb2[i] = bih2[i] + bhh2[i]; }

    const long long len = lengths[b];
    const float* mrow = mixed  + (long long)b * T_DIM * 2;
    const float* prow = proj   + (long long)b * T_DIM * 4;
    float*       arow = adjust + (long long)b * T_DIM * 2;

    float h10 = 0.f, h11 = 0.f, c10 = 0.f, c11 = 0.f;
    float h2 = 0.f, c2 = 0.f;

    for (int t = 0; t < T_DIM; ++t) {
        const float2 xv = *(const float2*)(mrow + t * 2);
        const float x0 = xv.x, x1 = xv.y;
        const float2 wv = *(const float2*)(prow + t * 4 + 2);
        const float w0 = wv.x, w1 = wv.y;

        // p1 gates (8 pre-activations, fully independent -> ILP)
        float z[8];
        #pragma unroll
        for (int g = 0; g < 8; ++g)
            z[g] = fmaf(wi1[g * 2], x0,
                   fmaf(wi1[g * 2 + 1], x1,
                   fmaf(wh1[g * 2], h10,
                   fmaf(wh1[g * 2 + 1], h11, bb1[g]))));
        const float i0 = sigf(z[0]), i1 = sigf(z[1]);
        const float f0 = sigf(z[2]), f1 = sigf(z[3]);
        const float g0 = tanh_fast(z[4]), g1 = tanh_fast(z[5]);
        const float o0 = sigf(z[6]), o1 = sigf(z[7]);
        c10 = fmaf(f0, c10, i0 * g0);
        c11 = fmaf(f1, c11, i1 * g1);
        h10 = o0 * tanh_fast(c10);
        h11 = o1 * tanh_fast(c11);

        // p2 gates (4 pre-activations)
        float zz[4];
        #pragma unroll
        for (int g = 0; g < 4; ++g)
            zz[g] = fmaf(wi2[g * 2], w0,
                    fmaf(wi2[g * 2 + 1], w1,
                    fmaf(wh2[g], h2, bb2[g])));
        const float ii = sigf(zz[0]), ff = sigf(zz[1]);
        const float gg = tanh_fast(zz[2]), oo = sigf(zz[3]);
        c2 = fmaf(ff, c2, ii * gg);
        h2 = oo * tanh_fast(c2);

        const bool valid = (long long)t < len;
        const float p10 = valid ? h10 : 0.f;
        const float p11 = valid ? h11 : 0.f;
        const float p2v = valid ? h2  : 0.f;
        float2 av;
        av.x = x0 - p10 + p2v;
        av.y = x1 - p11 + p2v;
        *(float2*)(arow + t * 2) = av;
    }
}

// ---------------------------------------------------------------------------
extern "C" void kernel_launch(void* const* d_in, const int* in_sizes, int n_in,
                              void* d_out, int out_size, void* d_ws, size_t ws_size,
                              hipStream_t stream) {
    const float*          x       = (const float*)d_in[0];
    const unsigned char*  worry   = (const unsigned char*)d_in[1];
    const long long*      lengths = (const long long*)d_in[2];
    const float* Wnw  = (const float*)d_in[3];
    const float* bnw  = (const float*)d_in[4];
    const float* Ww   = (const float*)d_in[5];
    const float* bw   = (const float*)d_in[6];
    const float* Wih1 = (const float*)d_in[7];
    const float* Whh1 = (const float*)d_in[8];
    const float* bih1 = (const float*)d_in[9];
    const float* bhh1 = (const float*)d_in[10];
    const float* Wih2 = (const float*)d_in[11];
    const float* Whh2 = (const float*)d_in[12];
    const float* bih2 = (const float*)d_in[13];
    const float* bhh2 = (const float*)d_in[14];

    float* out_nw  = (float*)d_out;                       // [B,T,2]
    float* out_adj = (float*)d_out + (size_t)R_DIM * 2;   // [B,T,2]
    float* proj    = (float*)d_ws;                        // [R,4] = 2MB scratch

    // A: 8192 tiles of 16 rows, 8 waves/block -> 1024 blocks
    proj_wmma_kernel<<<R_DIM / 16 / 8, 256, 0, stream>>>(x, Wnw, bnw, Ww, bw, proj);
    // B: masked mix -> noworry_det (output #1)
    mix_kernel<<<(R_DIM + 255) / 256, 256, 0, stream>>>(proj, worry, out_nw);
    // C: both LSTMs + adjust_det (output #2); 64 lanes = 64 batch chains
    lstm_kernel<<<1, 64, 0, stream>>>(out_nw, proj, lengths,
                                      Wih1, Whh1, bih1, bhh1,
                                      Wih2, Whh2, bih2, bhh2,
                                      out_adj);
}